// GATEncoder_11742440587935
// MI455X (gfx1250) — compile-verified
//
#include <hip/hip_runtime.h>
#include <hip/hip_bf16.h>

typedef __attribute__((ext_vector_type(2))) float v2f;
typedef __attribute__((ext_vector_type(4))) float v4f;
typedef __attribute__((ext_vector_type(8))) float v8f;

#define HEADS 4
#define CH 128            // HEADS*OUT_C == IN_C == 128
#define NEG_SLOPE 0.2f

// ---- monotone float<->u32 encoding for atomic max over signed floats ----
__device__ __forceinline__ unsigned enc_f32(float f) {
  unsigned u = __float_as_uint(f);
  return (f >= 0.0f) ? (u | 0x80000000u) : ~u;
}
__device__ __forceinline__ float dec_f32(unsigned u) {
  return (u & 0x80000000u) ? __uint_as_float(u & 0x7fffffffu)
                           : __uint_as_float(~u);
}

__device__ __forceinline__ void edge_nodes(const int* __restrict__ ei, int e,
                                           int ne, int& s, int& d) {
  if (e < ne) { s = ei[e]; d = ei[ne + e]; }
  else        { s = d = e - ne; }           // appended self-loops
}

// ---------------- init: zero out accumulators ----------------
__global__ void gat_init(float* __restrict__ out, unsigned* __restrict__ emax,
                         float* __restrict__ denom, int n) {
  int t = blockIdx.x * blockDim.x + threadIdx.x;
  if (t < n * CH) out[t] = 0.0f;
  if (t < n * HEADS) { emax[t] = 0u; denom[t] = 0.0f; }
}

// ---------------- h = x @ W via V_WMMA_F32_16X16X4_F32 ----------------
// Block = 256 threads (8 waves). Block b owns rows [16b,16b+16); wave w owns
// the 16-column tile starting at 16w. K loop: 32 steps of K=4.
__global__ __launch_bounds__(256) void gat_gemm_wmma(
    const float* __restrict__ x, const float* __restrict__ W,
    float* __restrict__ h, int n) {
  const int wave = threadIdx.x >> 5;      // 0..7 -> column tile
  const int lane = threadIdx.x & 31;
  const int m0 = blockIdx.x * 16;
  const int n0 = wave * 16;
  const int lm = lane & 15;               // M (for A) / N (for B)
  const int lk = (lane >> 4) * 2;         // K-pair select: 0 or 2

  int row = m0 + lm;
  if (row >= n) row = n - 1;              // clamp loads; stores are guarded
  const float* __restrict__ arow = x + (size_t)row * CH;

  v8f acc = {};
#pragma unroll
  for (int k0 = 0; k0 < CH; k0 += 4) {
    // A 16x4 f32: lane holds A[lm][k0+lk], A[lm][k0+lk+1]
    v2f a = *(const v2f*)(arow + k0 + lk);          // 8B aligned
    // B 4x16 f32: lane holds W[k0+lk][n0+lm], W[k0+lk+1][n0+lm]
    v2f b;
    b.x = W[(size_t)(k0 + lk) * CH + n0 + lm];
    b.y = W[(size_t)(k0 + lk + 1) * CH + n0 + lm];
    acc = __builtin_amdgcn_wmma_f32_16x16x4_f32(
        /*neg_a=*/false, a, /*neg_b=*/false, b,
        /*c_mod=*/(short)0, acc, /*reuse_a=*/false, /*reuse_b=*/false);
  }

  // C/D 16x16 f32 layout: col = n0 + (lane&15), row(v) = m0 + v + 8*(lane>>4)
  const int colg = n0 + lm;
  const int rbase = m0 + 8 * (lane >> 4);
#pragma unroll
  for (int v = 0; v < 8; ++v) {
    int rowg = rbase + v;
    if (rowg < n) h[(size_t)rowg * CH + colg] = acc[v];
  }
}

// ---------------- per-(node,head) attention dots ----------------
__global__ void gat_att(const float* __restrict__ h,
                        const float* __restrict__ att_src,
                        const float* __restrict__ att_dst,
                        float* __restrict__ a_src, float* __restrict__ a_dst,
                        int n) {
  int t = blockIdx.x * blockDim.x + threadIdx.x;   // node*HEADS + head
  if (t >= n * HEADS) return;
  int head = t & (HEADS - 1);
  const float* __restrict__ hv = h + (size_t)(t >> 2) * CH + head * 32;
  const float* __restrict__ as = att_src + head * 32;
  const float* __restrict__ ad = att_dst + head * 32;
  float s = 0.0f, d = 0.0f;
#pragma unroll
  for (int c = 0; c < 32; ++c) { float v = hv[c]; s += v * as[c]; d += v * ad[c]; }
  a_src[t] = s;
  a_dst[t] = d;
}

// ---------------- segment max over edges (atomic, encoded u32) ----------------
__global__ void gat_emax(const int* __restrict__ ei,
                         const float* __restrict__ a_src,
                         const float* __restrict__ a_dst,
                         unsigned* __restrict__ emax, int n, int ne) {
  int t = blockIdx.x * blockDim.x + threadIdx.x;   // (edge, head)
  if (t >= (ne + n) * HEADS) return;
  int head = t & (HEADS - 1);
  int e = t >> 2;
  int s, d; edge_nodes(ei, e, ne, s, d);
  float v = a_src[s * HEADS + head] + a_dst[d * HEADS + head];
  v = (v >= 0.0f) ? v : NEG_SLOPE * v;
  atomicMax(&emax[d * HEADS + head], enc_f32(v));
}

// ---------------- segment sum of exp(e - max) ----------------
__global__ void gat_denom(const int* __restrict__ ei,
                          const float* __restrict__ a_src,
                          const float* __restrict__ a_dst,
                          const unsigned* __restrict__ emax,
                          float* __restrict__ denom, int n, int ne) {
  int t = blockIdx.x * blockDim.x + threadIdx.x;   // (edge, head)
  if (t >= (ne + n) * HEADS) return;
  int head = t & (HEADS - 1);
  int e = t >> 2;
  int s, d; edge_nodes(ei, e, ne, s, d);
  float v = a_src[s * HEADS + head] + a_dst[d * HEADS + head];
  v = (v >= 0.0f) ? v : NEG_SLOPE * v;
  float m = dec_f32(emax[d * HEADS + head]);
  atomicAdd(&denom[d * HEADS + head], expf(v - m));
}

// ---------------- weighted scatter: out[dst] += alpha * h[src] ----------------
// One wave per edge; lane L handles features [4L, 4L+4) (single head per lane).
__global__ void gat_scatter(const int* __restrict__ ei,
                            const float* __restrict__ h,
                            const float* __restrict__ a_src,
                            const float* __restrict__ a_dst,
                            const unsigned* __restrict__ emax,
                            const float* __restrict__ denom,
                            float* __restrict__ out, int n, int ne) {
  int gid = blockIdx.x * blockDim.x + threadIdx.x;
  int e = gid >> 5;
  int lane = gid & 31;
  if (e >= ne + n) return;
  int s, d; edge_nodes(ei, e, ne, s, d);
  int f0 = lane * 4;
  int head = f0 >> 5;
  float v = a_src[s * HEADS + head] + a_dst[d * HEADS + head];
  v = (v >= 0.0f) ? v : NEG_SLOPE * v;
  float m = dec_f32(emax[d * HEADS + head]);
  float alpha = expf(v - m) / (denom[d * HEADS + head] + 1e-16f);
  v4f hv = *(const v4f*)(h + (size_t)s * CH + f0);   // 16B aligned
  float* __restrict__ od = out + (size_t)d * CH + f0;
  atomicAdd(od + 0, hv.x * alpha);
  atomicAdd(od + 1, hv.y * alpha);
  atomicAdd(od + 2, hv.z * alpha);
  atomicAdd(od + 3, hv.w * alpha);
}

// ---------------- bias + PReLU ----------------
__global__ void gat_prelu(float* __restrict__ out, const float* __restrict__ bias,
                          const float* __restrict__ prelu_w, int n) {
  int t = blockIdx.x * blockDim.x + threadIdx.x;
  if (t >= n * CH) return;
  float o = out[t] + bias[t & (CH - 1)];
  float pw = prelu_w[0];
  out[t] = (o >= 0.0f) ? o : pw * o;
}

extern "C" void kernel_launch(void* const* d_in, const int* in_sizes, int n_in,
                              void* d_out, int out_size, void* d_ws, size_t ws_size,
                              hipStream_t stream) {
  const float* x       = (const float*)d_in[0];
  const int*   ei      = (const int*)d_in[1];     // [2, E] flattened
  const float* W       = (const float*)d_in[2];
  const float* att_src = (const float*)d_in[3];
  const float* att_dst = (const float*)d_in[4];
  const float* bias    = (const float*)d_in[5];
  const float* prelu_w = (const float*)d_in[6];
  float* out = (float*)d_out;

  const int n  = in_sizes[0] / CH;   // nodes
  const int ne = in_sizes[1] / 2;    // edges

  // workspace layout
  float*    h      = (float*)d_ws;                          // n*128 f32
  float*    a_src  = h + (size_t)n * CH;                    // n*4
  float*    a_dst  = a_src + (size_t)n * HEADS;             // n*4
  unsigned* emax   = (unsigned*)(a_dst + (size_t)n * HEADS);// n*4
  float*    denom  = (float*)(emax + (size_t)n * HEADS);    // n*4

  const int tpb = 256;
  int total_eh = (ne + n) * HEADS;

  gat_init<<<(n * CH + tpb - 1) / tpb, tpb, 0, stream>>>(out, emax, denom, n);
  gat_gemm_wmma<<<(n + 15) / 16, 256, 0, stream>>>(x, W, h, n);
  gat_att<<<(n * HEADS + tpb - 1) / tpb, tpb, 0, stream>>>(h, att_src, att_dst,
                                                           a_src, a_dst, n);
  gat_emax<<<(total_eh + tpb - 1) / tpb, tpb, 0, stream>>>(ei, a_src, a_dst,
                                                           emax, n, ne);
  gat_denom<<<(total_eh + tpb - 1) / tpb, tpb, 0, stream>>>(ei, a_src, a_dst,
                                                            emax, denom, n, ne);
  long long sc_threads = (long long)(ne + n) * 32;
  gat_scatter<<<(int)((sc_threads + tpb - 1) / tpb), tpb, 0, stream>>>(
      ei, h, a_src, a_dst, emax, denom, out, n, ne);
  gat_prelu<<<(n * CH + tpb - 1) / tpb, tpb, 0, stream>>>(out, bias, prelu_w, n);
}